// GINEncoder_20401094656403
// MI455X (gfx1250) — compile-verified
//
#include <hip/hip_runtime.h>

// GIN encoder: agg = segment_sum(x[src], dst); h = (1+eps)x + agg;
// h = relu(bn0(h@W0+b0)); h = relu(bn1(h@W1+b1)); h = bn_o(relu(h@W2+b2));
// out = (h@Wm+bm, h@Wv+bv)

#define F_IN   128
#define HID    256
#define BN_EPS 1e-3f
#define MT     64     // node rows per workgroup
#define LDA    260    // LDS row stride (floats) for 256-wide activations (260%64==4 -> conflict-free)
#define LDIN   132    // LDS row stride (floats) for 128-wide input tile  (132%64==4 -> conflict-free)

typedef __attribute__((ext_vector_type(2))) float v2f;
typedef __attribute__((ext_vector_type(8))) float v8f;
typedef __attribute__((ext_vector_type(4))) unsigned int v4u;
typedef __attribute__((ext_vector_type(4))) int v4i;
typedef __attribute__((ext_vector_type(8))) int v8i;

#if defined(__gfx1250__) && __has_builtin(__builtin_amdgcn_tensor_load_to_lds)
#define USE_TDM 1
#else
#define USE_TDM 0
#endif

// ---------------- Kernel 1: h = (1+eps) * x ----------------
__global__ void gin_init_kernel(const float4* __restrict__ x4,
                                const float* __restrict__ eps,
                                float4* __restrict__ h4, int n4) {
  int i = blockIdx.x * blockDim.x + threadIdx.x;
  if (i >= n4) return;
  float s = 1.0f + eps[0];
  float4 v = x4[i];
  v.x *= s; v.y *= s; v.z *= s; v.w *= s;
  h4[i] = v;
}

// ---------------- Kernel 2: h[dst] += x[src] (edge scatter-add) ----------------
// One wave per edge; lane l owns features [4l, 4l+4). x row (512B) gather is
// fully coalesced; adds are L2-resident f32 atomics.
__global__ void gin_agg_kernel(const float* __restrict__ x,
                               const int* __restrict__ src,
                               const int* __restrict__ dst,
                               float* __restrict__ h, int E) {
  int idx = blockIdx.x * blockDim.x + threadIdx.x;
  int e    = idx >> 5;
  int lane = idx & 31;
  if (e >= E) return;
  int s = src[e];
  int d = dst[e];
  const float4 v = *(const float4*)(x + (size_t)s * F_IN + lane * 4);
  float* p = h + (size_t)d * F_IN + lane * 4;
  __hip_atomic_fetch_add(p + 0, v.x, __ATOMIC_RELAXED, __HIP_MEMORY_SCOPE_AGENT);
  __hip_atomic_fetch_add(p + 1, v.y, __ATOMIC_RELAXED, __HIP_MEMORY_SCOPE_AGENT);
  __hip_atomic_fetch_add(p + 2, v.z, __ATOMIC_RELAXED, __HIP_MEMORY_SCOPE_AGENT);
  __hip_atomic_fetch_add(p + 3, v.w, __ATOMIC_RELAXED, __HIP_MEMORY_SCOPE_AGENT);
}

// ---------------- WMMA f32 16x16x4 K-loop over an LDS tile ----------------
// A (16x4 f32): lanes 0-15 -> M=lane, K={0,1}; lanes 16-31 -> K={2,3}
// B (4x16 f32): lanes 0-15 -> N=lane, K={0,1}; lanes 16-31 -> K={2,3}
// C/D (16x16 f32, v8f): element r -> M = r + 8*(lane>>4), N = lane&15
__device__ inline v8f wmma_kloop(const float* lds, int ldl, int mrow,
                                 const float* __restrict__ W, int ncol,
                                 int K, int lm, int half) {
  v8f acc = {0.f, 0.f, 0.f, 0.f, 0.f, 0.f, 0.f, 0.f};
  for (int k = 0; k < K; k += 4) {
    int ka = k + 2 * half;
    const float* ap = lds + (mrow + lm) * ldl + ka;
    v2f a; a[0] = ap[0]; a[1] = ap[1];
    const float* bp = W + (size_t)ka * HID + ncol + lm;
    v2f b; b[0] = bp[0]; b[1] = bp[HID];
    acc = __builtin_amdgcn_wmma_f32_16x16x4_f32(false, a, false, b,
                                                (short)0, acc, false, false);
  }
  return acc;
}

// ---------------- Kernel 3: fused 5-layer MLP + BN + heads ----------------
__global__ __launch_bounds__(256) void gin_mlp_kernel(
    const float* __restrict__ h,
    const float* __restrict__ W0, const float* __restrict__ b0,
    const float* __restrict__ W1, const float* __restrict__ b1,
    const float* __restrict__ W2, const float* __restrict__ b2,
    const float* __restrict__ g0, const float* __restrict__ be0,
    const float* __restrict__ mm0, const float* __restrict__ mv0,
    const float* __restrict__ g1, const float* __restrict__ be1,
    const float* __restrict__ mm1, const float* __restrict__ mv1,
    const float* __restrict__ go, const float* __restrict__ beo,
    const float* __restrict__ mmo, const float* __restrict__ mvo,
    const float* __restrict__ Wm, const float* __restrict__ bm,
    const float* __restrict__ Wv, const float* __restrict__ bv,
    float* __restrict__ outM, float* __restrict__ outV, int N) {
  extern __shared__ float smem[];
  float* bufA  = smem;             // MT x LDA activations
  float* bufB  = smem + MT * LDA;  // MT x LDA activations
  float* bufIn = bufA;             // MT x LDIN input tile, overlaid on bufA

  const int tid  = threadIdx.x;
  const int wave = tid >> 5;
  const int lane = tid & 31;
  const int half = lane >> 4;
  const int lm   = lane & 15;
  const int row0 = blockIdx.x * MT;

#if USE_TDM
  // ---- stage in via Tensor Data Mover: MT x 128 f32 tile, LDS-padded to stride 132 ----
  // D# per CDNA5 ISA. tensor_dim1 = rows remaining
  // => hardware zero-fills the ragged last tile (OOB reads return zero).
  if (wave == 0) {
    unsigned long long ga = (unsigned long long)(size_t)(h + (size_t)row0 * F_IN);
    unsigned ldsAddr = (unsigned)(size_t)(__attribute__((address_space(3))) const void*)bufIn;
    unsigned rowsValid = (unsigned)(N - row0);
    v4u d0;
    d0[0] = 1u;                                           // count=1 (valid user D#)
    d0[1] = ldsAddr;                                      // lds_addr (bytes)
    d0[2] = (unsigned)(ga & 0xffffffffu);                 // global_addr[31:0]
    d0[3] = (unsigned)((ga >> 32) & 0x01ffffffu)          // global_addr[56:32]
            | (2u << 30);                                 // type=2 ("image")
    v8i d1;
    d1[0] = (int)((2u << 16)                              // data_size = 4B
                  | (1u << 20)                            // pad_enable
                  | (6u << 22)                            // pad_interval: 128 DWORDs
                  | (3u << 25));                          // pad_amount: 4 DWORDs -> stride 132
    d1[1] = (int)((unsigned)F_IN << 16);                  // tensor_dim0 = 128
    d1[2] = (int)((rowsValid & 0xffffu) << 16);           // tensor_dim1[15:0]
    d1[3] = (int)(((rowsValid >> 16) & 0xffffu)           // tensor_dim1[31:16]
                  | ((unsigned)F_IN << 16));              // tile_dim0 = 128
    d1[4] = (int)MT;                                      // tile_dim1 = 64 (tile_dim2 = 0)
    d1[5] = (int)F_IN;                                    // tensor_dim0_stride = 128
    d1[6] = 0;
    d1[7] = 0;
    v4i dz = {0, 0, 0, 0};
#if __clang_major__ >= 23
    v8i dz8 = {0, 0, 0, 0, 0, 0, 0, 0};
    __builtin_amdgcn_tensor_load_to_lds(d0, d1, dz, dz, dz8, 0);
#else
    __builtin_amdgcn_tensor_load_to_lds(d0, d1, dz, dz, 0);
#endif
    __builtin_amdgcn_s_wait_tensorcnt(0);
  }
  __syncthreads();
#else
  // ---- fallback stage-in: cooperative copy (also used for host pass) ----
  for (int i = tid; i < MT * (F_IN / 4); i += 256) {
    int r  = i >> 5;          // F_IN/4 == 32 float4 per row
    int c4 = i & 31;
    float4 v = make_float4(0.f, 0.f, 0.f, 0.f);
    int gr = row0 + r;
    if (gr < N) v = *(const float4*)(h + (size_t)gr * F_IN + c4 * 4);
    float* p = bufIn + r * LDIN + c4 * 4;
    p[0] = v.x; p[1] = v.y; p[2] = v.z; p[3] = v.w;
  }
  __syncthreads();
#endif

  const int NT = (MT / 16) * (HID / 16);  // 64 output tiles per layer

  // ---- layer 0: bufIn(K=128) @ W0 -> bn0 -> relu -> bufB ----
  for (int t = wave; t < NT; t += 8) {
    int mrow = (t >> 4) << 4, ncol = (t & 15) << 4;
    v8f acc = wmma_kloop(bufIn, LDIN, mrow, W0, ncol, F_IN, lm, half);
    int c = ncol + lm;
    float sc = g0[c] * rsqrtf(mv0[c] + BN_EPS);
    float sh = be0[c] - mm0[c] * sc;
    float bb = b0[c];
#pragma unroll
    for (int r = 0; r < 8; ++r) {
      float v = acc[r] + bb;
      bufB[(mrow + r + 8 * half) * LDA + c] = fmaxf(v * sc + sh, 0.f);
    }
  }
  __syncthreads();

  // ---- layer 1: bufB(K=256) @ W1 -> bn1 -> relu -> bufA ----
  for (int t = wave; t < NT; t += 8) {
    int mrow = (t >> 4) << 4, ncol = (t & 15) << 4;
    v8f acc = wmma_kloop(bufB, LDA, mrow, W1, ncol, HID, lm, half);
    int c = ncol + lm;
    float sc = g1[c] * rsqrtf(mv1[c] + BN_EPS);
    float sh = be1[c] - mm1[c] * sc;
    float bb = b1[c];
#pragma unroll
    for (int r = 0; r < 8; ++r) {
      float v = acc[r] + bb;
      bufA[(mrow + r + 8 * half) * LDA + c] = fmaxf(v * sc + sh, 0.f);
    }
  }
  __syncthreads();

  // ---- layer 2: bufA(K=256) @ W2 -> relu -> outer bn -> bufB ----
  for (int t = wave; t < NT; t += 8) {
    int mrow = (t >> 4) << 4, ncol = (t & 15) << 4;
    v8f acc = wmma_kloop(bufA, LDA, mrow, W2, ncol, HID, lm, half);
    int c = ncol + lm;
    float sc = go[c] * rsqrtf(mvo[c] + BN_EPS);
    float sh = beo[c] - mmo[c] * sc;
    float bb = b2[c];
#pragma unroll
    for (int r = 0; r < 8; ++r) {
      float v = fmaxf(acc[r] + bb, 0.f);  // relu BEFORE outer BN, per reference
      bufB[(mrow + r + 8 * half) * LDA + c] = v * sc + sh;
    }
  }
  __syncthreads();

  // ---- heads: bufB(K=256) @ Wm + bm -> outM ; @ Wv + bv -> outV ----
  for (int t = wave; t < NT; t += 8) {
    int mrow = (t >> 4) << 4, ncol = (t & 15) << 4;
    int c = ncol + lm;
    v8f am = wmma_kloop(bufB, LDA, mrow, Wm, ncol, HID, lm, half);
    float bbm = bm[c];
#pragma unroll
    for (int r = 0; r < 8; ++r) {
      int gr = row0 + mrow + r + 8 * half;
      if (gr < N) outM[(size_t)gr * HID + c] = am[r] + bbm;
    }
    v8f av = wmma_kloop(bufB, LDA, mrow, Wv, ncol, HID, lm, half);
    float bbv = bv[c];
#pragma unroll
    for (int r = 0; r < 8; ++r) {
      int gr = row0 + mrow + r + 8 * half;
      if (gr < N) outV[(size_t)gr * HID + c] = av[r] + bbv;
    }
  }
}

extern "C" void kernel_launch(void* const* d_in, const int* in_sizes, int n_in,
                              void* d_out, int out_size, void* d_ws, size_t ws_size,
                              hipStream_t stream) {
  const float* x   = (const float*)d_in[0];
  const int*   src = (const int*)d_in[1];
  const int*   dst = (const int*)d_in[2];
  const float* eps = (const float*)d_in[3];
  const float* W0  = (const float*)d_in[4];
  const float* b0  = (const float*)d_in[5];
  const float* W1  = (const float*)d_in[6];
  const float* b1  = (const float*)d_in[7];
  const float* W2  = (const float*)d_in[8];
  const float* b2  = (const float*)d_in[9];
  const float* g0  = (const float*)d_in[10];
  const float* be0 = (const float*)d_in[11];
  const float* mm0 = (const float*)d_in[12];
  const float* mv0 = (const float*)d_in[13];
  const float* g1  = (const float*)d_in[14];
  const float* be1 = (const float*)d_in[15];
  const float* mm1 = (const float*)d_in[16];
  const float* mv1 = (const float*)d_in[17];
  const float* go  = (const float*)d_in[18];
  const float* beo = (const float*)d_in[19];
  const float* mmo = (const float*)d_in[20];
  const float* mvo = (const float*)d_in[21];
  const float* Wm  = (const float*)d_in[22];
  const float* bm  = (const float*)d_in[23];
  const float* Wv  = (const float*)d_in[24];
  const float* bv  = (const float*)d_in[25];

  const int N = in_sizes[0] / F_IN;
  const int E = in_sizes[1];
  float* h = (float*)d_ws;  // [N, 128] aggregated features

  // 1) h = (1+eps) * x
  int n4 = N * (F_IN / 4);
  gin_init_kernel<<<(n4 + 255) / 256, 256, 0, stream>>>(
      (const float4*)x, eps, (float4*)h, n4);

  // 2) h[dst] += x[src]
  long long tot = (long long)E * 32;
  gin_agg_kernel<<<(unsigned)((tot + 255) / 256), 256, 0, stream>>>(x, src, dst, h, E);

  // 3) fused MLP + BN + heads
  int blocks = (N + MT - 1) / MT;
  size_t shmem = (size_t)2 * MT * LDA * sizeof(float);  // 133,120 B (2 WGs fit in 320 KB WGP LDS)
  gin_mlp_kernel<<<blocks, 256, shmem, stream>>>(
      h, W0, b0, W1, b1, W2, b2,
      g0, be0, mm0, mv0, g1, be1, mm1, mv1, go, beo, mmo, mvo,
      Wm, bm, Wv, bv,
      (float*)d_out, (float*)d_out + (size_t)N * HID, N);
}